// GCN_Full_58909771432681
// MI455X (gfx1250) — compile-verified
//
#include <hip/hip_runtime.h>

#define NNODES 100000
#define NEDGES 1600000
#define FEATS  64

typedef __attribute__((ext_vector_type(2))) float v2f;
typedef __attribute__((ext_vector_type(4))) float v4f;
typedef __attribute__((ext_vector_type(8))) float v8f;

// ---------------- utility: zero fill (float4 granularity) ----------------
__global__ void zero_f4(float* __restrict__ p, int n4) {
    int i = blockIdx.x * blockDim.x + threadIdx.x;
    if (i < n4) {
        v4f z = {};
        ((v4f*)p)[i] = z;
    }
}

// ---------------- degree accumulation: deg[0..N)=out-deg, deg[N..2N)=in-deg
__global__ void degree_kernel(const int* __restrict__ src, const int* __restrict__ dst,
                              float* __restrict__ deg, int nE) {
    int e = blockIdx.x * blockDim.x + threadIdx.x;
    if (e < nE) {
        atomicAdd(&deg[src[e]], 1.0f);
        atomicAdd(&deg[NNODES + dst[e]], 1.0f);
    }
}

// ---------------- deg -> rsqrt(max(deg,1)) in place ----------------
__global__ void norm_kernel(float* __restrict__ deg, int n) {
    int i = blockIdx.x * blockDim.x + threadIdx.x;
    if (i < n) deg[i] = rsqrtf(fmaxf(deg[i], 1.0f));
}

// ---------------- edge gather + atomic scatter-add --------------------------
// 16 threads per edge, 4 feats per thread (float4 gather, 4 f32 atomics).
// agg (25.6 MB) is L2-resident on MI455X (192 MB L2), so atomics resolve in L2.
__global__ void gather_kernel(const float* __restrict__ h, const int* __restrict__ src,
                              const int* __restrict__ dst, const float* __restrict__ src_norm,
                              float* __restrict__ agg, int nE) {
    int idx = blockIdx.x * blockDim.x + threadIdx.x;
    int e = idx >> 4;
    if (e >= nE) return;
    int fo = (idx & 15) << 2;
    int s = src[e];
    int d = dst[e];
    float sn = src_norm[s];
    v4f v = *(const v4f*)(h + (size_t)s * FEATS + fo);
    float* ap = agg + (size_t)d * FEATS + fo;
    atomicAdd(ap + 0, v.x * sn);
    atomicAdd(ap + 1, v.y * sn);
    atomicAdd(ap + 2, v.z * sn);
    atomicAdd(ap + 3, v.w * sn);
}

// ---------------- fused (agg * dst_norm) @ W + b (+relu) via f32 WMMA -------
// One wave per 16-row block; NT column tiles of 16 (NOUT = NT*16).
// V_WMMA_F32_16X16X4_F32 operand layouts per CDNA5 ISA 7.12.2:
//   A 16x4:  lanes 0-15 -> K = {k0, k0+1}, lanes 16-31 -> K = {k0+2, k0+3}
//   B 4x16:  same K split across lane halves, N = lane%16
//   C/D:     VGPR i -> row base+i (lanes 0-15), row base+i+8 (lanes 16-31)
template <int NT, bool RELU>
__global__ void gemm_wmma(const float* __restrict__ agg, const float* __restrict__ dst_norm,
                          const float* __restrict__ W, const float* __restrict__ bias,
                          float* __restrict__ out, int nRows) {
    constexpr int NOUT = NT * 16;
    const int lane = threadIdx.x & 31;
    const int wave = threadIdx.x >> 5;
    const int rowBase = (blockIdx.x * (blockDim.x >> 5) + wave) * 16;
    if (rowBase >= nRows) return;  // wave-uniform: EXEC stays all-ones for WMMA

    const int half = lane >> 4;   // 0: low K pair / rows 0-7, 1: high K pair / rows 8-15
    const int l16  = lane & 15;

    int arow = rowBase + l16;
    if (arow >= nRows) arow = nRows - 1;        // clamp (nRows % 16 == 0 in practice)
    const float dn = dst_norm[arow];
    const float* aptr = agg + (size_t)arow * FEATS + half * 2;

    v8f acc[NT] = {};

#pragma unroll
    for (int k0 = 0; k0 < FEATS; k0 += 4) {
        v2f A;
        A.x = aptr[k0 + 0] * dn;                 // fuse dst_norm row scaling into A
        A.y = aptr[k0 + 1] * dn;
        const float* wp = W + (size_t)(k0 + half * 2) * NOUT + l16;
#pragma unroll
        for (int t = 0; t < NT; ++t) {
            v2f B;
            B.x = wp[t * 16];
            B.y = wp[t * 16 + NOUT];
            acc[t] = __builtin_amdgcn_wmma_f32_16x16x4_f32(
                false, A, false, B, (short)0, acc[t], false, false);
        }
    }

#pragma unroll
    for (int t = 0; t < NT; ++t) {
        const int col = t * 16 + l16;
        const float bv = bias[col];
#pragma unroll
        for (int i = 0; i < 8; ++i) {
            const int row = rowBase + half * 8 + i;
            float v = acc[t][i] + bv;
            if (RELU) v = fmaxf(v, 0.0f);
            if (row < nRows) out[(size_t)row * NOUT + col] = v;
        }
    }
}

extern "C" void kernel_launch(void* const* d_in, const int* in_sizes, int n_in,
                              void* d_out, int out_size, void* d_ws, size_t ws_size,
                              hipStream_t stream) {
    const float* x    = (const float*)d_in[0];
    const int*   esrc = (const int*)d_in[1];
    const int*   edst = (const int*)d_in[2];
    const float* W0   = (const float*)d_in[3];
    const float* b0   = (const float*)d_in[4];
    const float* W1   = (const float*)d_in[5];
    const float* b1   = (const float*)d_in[6];
    const float* W2   = (const float*)d_in[7];
    const float* b2   = (const float*)d_in[8];
    float* outp = (float*)d_out;

    // workspace layout (floats): [src_norm N][dst_norm N][agg N*64][h N*64]
    float* ws       = (float*)d_ws;
    float* norms    = ws;                              // 2N (deg -> norm in place)
    float* src_norm = norms;
    float* dst_norm = norms + NNODES;
    float* agg      = norms + 2 * NNODES;
    float* h        = agg + (size_t)NNODES * FEATS;

    const int T = 256;
    const int n4feat   = NNODES * FEATS / 4;
    const int gZeroN   = (2 * NNODES / 4 + T - 1) / T;
    const int gZeroF   = (n4feat + T - 1) / T;
    const int gEdge    = (NEDGES + T - 1) / T;
    const int gGather  = (NEDGES * 16 + T - 1) / T;
    const int wavesPerBlock = T / 32;
    const int gGemm    = (NNODES + 16 * wavesPerBlock - 1) / (16 * wavesPerBlock);

    // degrees -> norms
    zero_f4<<<gZeroN, T, 0, stream>>>(norms, 2 * NNODES / 4);
    degree_kernel<<<gEdge, T, 0, stream>>>(esrc, edst, norms, NEDGES);
    norm_kernel<<<(2 * NNODES + T - 1) / T, T, 0, stream>>>(norms, 2 * NNODES);

    // layer 0: x -> h   (relu)
    zero_f4<<<gZeroF, T, 0, stream>>>(agg, n4feat);
    gather_kernel<<<gGather, T, 0, stream>>>(x, esrc, edst, src_norm, agg, NEDGES);
    gemm_wmma<4, true><<<gGemm, T, 0, stream>>>(agg, dst_norm, W0, b0, h, NNODES);

    // layer 1: h -> h   (relu)
    zero_f4<<<gZeroF, T, 0, stream>>>(agg, n4feat);
    gather_kernel<<<gGather, T, 0, stream>>>(h, esrc, edst, src_norm, agg, NEDGES);
    gemm_wmma<4, true><<<gGemm, T, 0, stream>>>(agg, dst_norm, W1, b1, h, NNODES);

    // layer 2: h -> out (no relu, 32 classes)
    zero_f4<<<gZeroF, T, 0, stream>>>(agg, n4feat);
    gather_kernel<<<gGather, T, 0, stream>>>(h, esrc, edst, src_norm, agg, NEDGES);
    gemm_wmma<2, false><<<gGemm, T, 0, stream>>>(agg, dst_norm, W2, b2, outp, NNODES);
}